// GCNModel_38920993636608
// MI455X (gfx1250) — compile-verified
//
#include <hip/hip_runtime.h>
#include <math.h>

typedef float v2f __attribute__((ext_vector_type(2)));
typedef float v8f __attribute__((ext_vector_type(8)));

#define BN_EPS 1e-5f

// GEMM tiling
#define BM 64          // rows per block
#define BKC 32         // K chunk
#define A_LDST 36      // LDS row stride for A (pad: conflict-free, 16B aligned)
#define B_LDST 136     // LDS row stride for B (pad: conflict-free, 16B aligned)

#if __has_builtin(__builtin_amdgcn_global_load_async_to_lds_b128)
#define GCN_ASYNC_LDS 1
#endif

#ifdef GCN_ASYNC_LDS
typedef int b128 __attribute__((vector_size(16)));
typedef __attribute__((address_space(1))) b128 b128_g;   // global (AS1)
typedef __attribute__((address_space(3))) b128 b128_l;   // LDS (AS3)

// GLOBAL_LOAD_ASYNC_TO_LDS_B128: VGPR-free DMA, tracked by ASYNCcnt.
// AS1 value == flat address; AS3 value == low 32 bits of flat shared address.
__device__ __forceinline__ void async_cp16(const float* g, const float* l) {
    __builtin_amdgcn_global_load_async_to_lds_b128(
        (b128_g*)(uintptr_t)g,
        (b128_l*)(uintptr_t)(unsigned)(uintptr_t)l,
        0, 0);
}
__device__ __forceinline__ void wait_async0() {
#if __has_builtin(__builtin_amdgcn_s_wait_asynccnt)
    __builtin_amdgcn_s_wait_asynccnt(0);
#else
    asm volatile("s_wait_asynccnt 0x0" ::: "memory");
#endif
}
#endif

// ---------------------------------------------------------------- utility
__global__ __launch_bounds__(256) void k_fill(float* __restrict__ p, float v, int n) {
    int i = blockIdx.x * blockDim.x + threadIdx.x;
    if (i < n) p[i] = v;
}

__global__ __launch_bounds__(256) void k_count_deg(const int* __restrict__ dst,
                                                   float* __restrict__ deg, int E) {
    int i = blockIdx.x * blockDim.x + threadIdx.x;
    if (i < E) atomicAdd(&deg[dst[i]], 1.0f);
}

__global__ __launch_bounds__(256) void k_rsqrt_inplace(float* __restrict__ p, int n) {
    int i = blockIdx.x * blockDim.x + threadIdx.x;
    if (i < n) p[i] = rsqrtf(p[i]);   // deg >= 1 always (self-loop)
}

// ---------------------------------------------------------------- fp32 WMMA GEMM
// C[M,N] = A[M,K] @ B[K,N]; row-major; requires N == 128, K % BKC == 0.
// Block (32,8) = 8 waves: 64x128 C tile; wave -> 16 cols x 4 M-subtiles.
// Double-buffered LDS staging via GLOBAL_LOAD_ASYNC_TO_LDS_B128 (if available).
__global__ __launch_bounds__(256) void k_gemm_wmma_f32(const float* __restrict__ A,
                                                       const float* __restrict__ B,
                                                       float* __restrict__ C,
                                                       int M, int K, int N) {
    __shared__ float As[2][BM * A_LDST];    // 2 x  9216 B
    __shared__ float Bs[2][BKC * B_LDST];   // 2 x 17408 B

    const int tid    = threadIdx.y * 32 + threadIdx.x;  // 0..255
    const int lane   = threadIdx.x;                     // 0..31
    const int hi     = lane >> 4;                       // K sub-pair / C row group
    const int ln16   = lane & 15;                       // M row (A) / N col (B,C)
    const int nbase  = threadIdx.y * 16;
    const int mblock = blockIdx.x * BM;

    // global->LDS staging coordinates
    const int ar = tid >> 2;          // A row 0..63 (4 threads/row)
    const int ac = (tid & 3) * 8;     // A col 0,8,16,24
    int arow = mblock + ar;
    if (arow >= M) arow = M - 1;      // clamp tail rows (stores are guarded)
    const float* agp = A + (size_t)arow * K + ac;
    const int br = tid >> 3;          // B row 0..31 (8 threads/row)
    const int bc = (tid & 7) * 16;    // B col 0..112

    v8f acc[4] = {};

#ifdef GCN_ASYNC_LDS
    auto issue = [&](int buf, int k0) {
        const float* ag = agp + k0;
        const float* al = &As[buf][ar * A_LDST + ac];
        async_cp16(ag, al);
        async_cp16(ag + 4, al + 4);
        const float* bg = B + (size_t)(k0 + br) * N + bc;
        const float* bl = &Bs[buf][br * B_LDST + bc];
        async_cp16(bg,      bl);
        async_cp16(bg + 4,  bl + 4);
        async_cp16(bg + 8,  bl + 8);
        async_cp16(bg + 12, bl + 12);
    };
    issue(0, 0);
    for (int k0 = 0; k0 < K; k0 += BKC) {
        const int cur = (k0 / BKC) & 1;
        wait_async0();                 // this wave's copies for chunk k0 landed
        __syncthreads();               // ... and everyone else's
        if (k0 + BKC < K) issue(cur ^ 1, k0 + BKC);  // DMA next chunk during compute
        #pragma unroll
        for (int k = 0; k < BKC; k += 4) {
            v2f b;
            b.x = Bs[cur][(k + 2 * hi)     * B_LDST + nbase + ln16];
            b.y = Bs[cur][(k + 2 * hi + 1) * B_LDST + nbase + ln16];
            #pragma unroll
            for (int mt = 0; mt < 4; ++mt) {
                v2f a = *(const v2f*)&As[cur][(mt * 16 + ln16) * A_LDST + k + 2 * hi];
                acc[mt] = __builtin_amdgcn_wmma_f32_16x16x4_f32(
                              false, a, false, b, (short)0, acc[mt], false, false);
            }
        }
        __syncthreads();               // readers done before buffer is re-filled
    }
#else
    for (int k0 = 0; k0 < K; k0 += BKC) {
        float4 a0 = *(const float4*)(agp + k0);
        float4 a1 = *(const float4*)(agp + k0 + 4);
        *(float4*)&As[0][ar * A_LDST + ac]     = a0;
        *(float4*)&As[0][ar * A_LDST + ac + 4] = a1;
        const float* bgp = B + (size_t)(k0 + br) * N + bc;
        float4 b0 = *(const float4*)(bgp + 0);
        float4 b1 = *(const float4*)(bgp + 4);
        float4 b2 = *(const float4*)(bgp + 8);
        float4 b3 = *(const float4*)(bgp + 12);
        float* bsp = &Bs[0][br * B_LDST + bc];
        *(float4*)(bsp + 0)  = b0;
        *(float4*)(bsp + 4)  = b1;
        *(float4*)(bsp + 8)  = b2;
        *(float4*)(bsp + 12) = b3;
        __syncthreads();
        if (k0 + BKC < K)
            __builtin_prefetch(agp + k0 + BKC, 0, 3);
        #pragma unroll
        for (int k = 0; k < BKC; k += 4) {
            v2f b;
            b.x = Bs[0][(k + 2 * hi)     * B_LDST + nbase + ln16];
            b.y = Bs[0][(k + 2 * hi + 1) * B_LDST + nbase + ln16];
            #pragma unroll
            for (int mt = 0; mt < 4; ++mt) {
                v2f a = *(const v2f*)&As[0][(mt * 16 + ln16) * A_LDST + k + 2 * hi];
                acc[mt] = __builtin_amdgcn_wmma_f32_16x16x4_f32(
                              false, a, false, b, (short)0, acc[mt], false, false);
            }
        }
        __syncthreads();
    }
#endif

    #pragma unroll
    for (int mt = 0; mt < 4; ++mt) {
        #pragma unroll
        for (int r = 0; r < 8; ++r) {
            int row = mblock + mt * 16 + r + 8 * hi;
            if (row < M) C[(size_t)row * N + nbase + ln16] = acc[mt][r];
        }
    }
}

// small-N GEMM (final layer, N=2)
__global__ __launch_bounds__(256) void k_gemm_small(const float* __restrict__ A,
                                                    const float* __restrict__ B,
                                                    float* __restrict__ C,
                                                    int M, int K, int N) {
    int i = blockIdx.x * blockDim.x + threadIdx.x;
    if (i >= M * N) return;
    int row = i / N, col = i % N;
    const float* a = A + (size_t)row * K;
    float s = 0.f;
    #pragma unroll 8
    for (int k = 0; k < K; ++k) s += a[k] * B[(size_t)k * N + col];
    C[i] = s;
}

// ---------------------------------------------------------------- aggregation
// out[i,f] = bias[f] + dis[i]^2 * msg[i,f]   (self-loop term + bias)
__global__ __launch_bounds__(256) void k_init_agg(const float* __restrict__ msg,
                                                  const float* __restrict__ dis,
                                                  const float* __restrict__ bias,
                                                  float* __restrict__ out,
                                                  int Mn, int F) {
    size_t i = (size_t)blockIdx.x * blockDim.x + threadIdx.x;
    if (i >= (size_t)Mn * F) return;
    int node = (int)(i / F), f = (int)(i % F);
    float d = dis[node];
    out[i] = bias[f] + d * d * msg[i];
}

// F==128: one wave per edge; lane handles 4 features (float4 gather, f32 atomics)
__global__ __launch_bounds__(256) void k_scatter128(const int* __restrict__ src,
                                                    const int* __restrict__ dst,
                                                    const float* __restrict__ dis,
                                                    const float* __restrict__ msg,
                                                    float* __restrict__ out, int E) {
    size_t tid = (size_t)blockIdx.x * blockDim.x + threadIdx.x;
    int e = (int)(tid >> 5);
    if (e >= E) return;
    int f = (int)(tid & 31) << 2;
    int s = src[e], d = dst[e];
    float nrm = dis[s] * dis[d];
    const float4 m4 = *(const float4*)(msg + ((size_t)s << 7) + f);
    float* o = out + ((size_t)d << 7) + f;
    atomicAdd(o + 0, nrm * m4.x);
    atomicAdd(o + 1, nrm * m4.y);
    atomicAdd(o + 2, nrm * m4.z);
    atomicAdd(o + 3, nrm * m4.w);
}

// generic small-F scatter (final layer, F=2)
__global__ __launch_bounds__(256) void k_scatter_f(const int* __restrict__ src,
                                                   const int* __restrict__ dst,
                                                   const float* __restrict__ dis,
                                                   const float* __restrict__ msg,
                                                   float* __restrict__ out,
                                                   int E, int F) {
    int e = blockIdx.x * blockDim.x + threadIdx.x;
    if (e >= E) return;
    int s = src[e], d = dst[e];
    float nrm = dis[s] * dis[d];
    for (int f = 0; f < F; ++f)
        atomicAdd(&out[(size_t)d * F + f], nrm * msg[(size_t)s * F + f]);
}

// ---------------------------------------------------------------- epilogues
__global__ __launch_bounds__(256) void k_bn_relu(float* __restrict__ h,
                                                 const float* __restrict__ g,
                                                 const float* __restrict__ be,
                                                 const float* __restrict__ m,
                                                 const float* __restrict__ v,
                                                 int Mn, int F) {
    size_t i = (size_t)blockIdx.x * blockDim.x + threadIdx.x;
    if (i >= (size_t)Mn * F) return;
    int f = (int)(i % F);
    float y = (h[i] - m[f]) * (g[f] * rsqrtf(v[f] + BN_EPS)) + be[f];
    h[i] = y > 0.f ? y : 0.f;
}

__global__ __launch_bounds__(256) void k_logsoftmax(float* __restrict__ out, int Mn, int F) {
    int i = blockIdx.x * blockDim.x + threadIdx.x;
    if (i >= Mn) return;
    float* r = out + (size_t)i * F;
    float mx = r[0];
    for (int f = 1; f < F; ++f) mx = fmaxf(mx, r[f]);
    float s = 0.f;
    for (int f = 0; f < F; ++f) s += __expf(r[f] - mx);
    float lse = mx + __logf(s);
    for (int f = 0; f < F; ++f) r[f] = r[f] - lse;
}

// ---------------------------------------------------------------- driver
extern "C" void kernel_launch(void* const* d_in, const int* in_sizes, int n_in,
                              void* d_out, int out_size, void* d_ws, size_t ws_size,
                              hipStream_t stream) {
    const float* x  = (const float*)d_in[0];
    const int*   ei = (const int*)d_in[1];
    const float* W1 = (const float*)d_in[2];  const float* b1 = (const float*)d_in[3];
    const float* W2 = (const float*)d_in[4];  const float* b2 = (const float*)d_in[5];
    const float* W3 = (const float*)d_in[6];  const float* b3 = (const float*)d_in[7];
    const float* W4 = (const float*)d_in[8];  const float* b4 = (const float*)d_in[9];
    const float* g1 = (const float*)d_in[10]; const float* be1 = (const float*)d_in[11];
    const float* m1 = (const float*)d_in[12]; const float* v1  = (const float*)d_in[13];
    const float* g2 = (const float*)d_in[14]; const float* be2 = (const float*)d_in[15];
    const float* m2 = (const float*)d_in[16]; const float* v2  = (const float*)d_in[17];
    const float* g3 = (const float*)d_in[18]; const float* be3 = (const float*)d_in[19];
    const float* m3 = (const float*)d_in[20]; const float* v3  = (const float*)d_in[21];

    const int H    = in_sizes[3];        // 128
    const int DIN  = in_sizes[2] / H;    // 512
    const int DOUT = in_sizes[9];        // 2
    const int Mn   = in_sizes[0] / DIN;  // 50000
    const int E    = in_sizes[1] / 2;    // 800000

    const int* src = ei;
    const int* dst = ei + E;

    float* ws   = (float*)d_ws;
    size_t nAl  = ((size_t)Mn + 255) & ~(size_t)255;
    float* dis  = ws;                    // N (deg then rsqrt(deg))
    float* buf0 = ws + nAl;              // N x H (messages)
    float* buf1 = buf0 + (size_t)Mn * H; // N x H (aggregated / activations)

    auto cdiv = [](long long a, long long b) { return (int)((a + b - 1) / b); };

    // symmetric normalization: deg counted on dst, self-loops included
    k_fill<<<cdiv(Mn, 256), 256, 0, stream>>>(dis, 1.0f, Mn);
    k_count_deg<<<cdiv(E, 256), 256, 0, stream>>>(dst, dis, E);
    k_rsqrt_inplace<<<cdiv(Mn, 256), 256, 0, stream>>>(dis, Mn);

    auto conv128 = [&](const float* hin, const float* W, const float* bias, int K) {
        k_gemm_wmma_f32<<<dim3(cdiv(Mn, BM), 1, 1), dim3(32, 8, 1), 0, stream>>>(
            hin, W, buf0, Mn, K, H);
        k_init_agg<<<cdiv((long long)Mn * H, 256), 256, 0, stream>>>(
            buf0, dis, bias, buf1, Mn, H);
        k_scatter128<<<cdiv((long long)E * 32, 256), 256, 0, stream>>>(
            src, dst, dis, buf0, buf1, E);
    };

    conv128(x, W1, b1, DIN);
    k_bn_relu<<<cdiv((long long)Mn * H, 256), 256, 0, stream>>>(buf1, g1, be1, m1, v1, Mn, H);
    conv128(buf1, W2, b2, H);
    k_bn_relu<<<cdiv((long long)Mn * H, 256), 256, 0, stream>>>(buf1, g2, be2, m2, v2, Mn, H);
    conv128(buf1, W3, b3, H);
    k_bn_relu<<<cdiv((long long)Mn * H, 256), 256, 0, stream>>>(buf1, g3, be3, m3, v3, Mn, H);

    // final conv (H -> DOUT) into d_out, then log_softmax in place
    float* out = (float*)d_out;
    k_gemm_small<<<cdiv((long long)Mn * DOUT, 256), 256, 0, stream>>>(buf1, W4, buf0, Mn, H, DOUT);
    k_init_agg<<<cdiv((long long)Mn * DOUT, 256), 256, 0, stream>>>(buf0, dis, b4, out, Mn, DOUT);
    k_scatter_f<<<cdiv(E, 256), 256, 0, stream>>>(src, dst, dis, buf0, out, E, DOUT);
    k_logsoftmax<<<cdiv(Mn, 256), 256, 0, stream>>>(out, Mn, DOUT);
}